// loss_CRPS_78666620994050
// MI455X (gfx1250) — compile-verified
//
#include <hip/hip_runtime.h>
#include <hip/hip_bf16.h>

// CRPS: samples [N=16, B*C*H*W = 262144] f32, target [262144] f32 -> scalar f32.
// Pass 1: 256 blocks x 256 threads, each thread owns 4 consecutive pixels
//         (float4 / b128 loads), computes term1 & pairwise term in registers,
//         reduces per-wave with V_WMMA_F32_16X16X4_F32 (B = ones), per-block
//         via LDS, writes 256 partials to d_ws.
// Pass 2: one wave32 reduces the 256 partials with one WMMA and writes d_out[0].

typedef float v2f __attribute__((ext_vector_type(2)));
typedef float v8f __attribute__((ext_vector_type(8)));

#define NSAMP    16
#define NPIX     262144            // 4*1*256*256
#define VEC      4
#define NTHREADS 256
#define NBLOCKS  ((NPIX / VEC) / NTHREADS)   // 256

// Sum 64 values spread as (a0, a1) across the 32 lanes of a wave using the
// matrix pipe: A is the 16x4 f32 WMMA operand (lane m: K0=a0,K1=a1; lane m+16:
// K2=a0,K3=a1), B = all-ones 4x16 -> D[m][n] = row-sum of A. Summing the 8 D
// VGPRs gives per-lane column partials; column 0's total = readlane 0 + 16.
// Requires EXEC == all ones (call with no divergence; full wave).
__device__ __forceinline__ float wave_sum64_wmma(float a0, float a1) {
  v2f A; A.x = a0; A.y = a1;
  v2f B; B.x = 1.0f; B.y = 1.0f;
  v8f C = {};
  v8f D = __builtin_amdgcn_wmma_f32_16x16x4_f32(
      /*neg_a=*/false, A, /*neg_b=*/false, B,
      /*c_mod=*/(short)0, C, /*reuse_a=*/false, /*reuse_b=*/false);
  float s = D[0] + D[1] + D[2] + D[3] + D[4] + D[5] + D[6] + D[7];
  return __int_as_float(__builtin_amdgcn_readlane(__float_as_int(s), 0)) +
         __int_as_float(__builtin_amdgcn_readlane(__float_as_int(s), 16));
}

__global__ void crps_partial_kernel(const float* __restrict__ samples,
                                    const float* __restrict__ target,
                                    float* __restrict__ partials) {
  const int t = blockIdx.x * NTHREADS + threadIdx.x;   // 0 .. 65535 (float4 idx)

  float4 y = reinterpret_cast<const float4*>(target)[t];

  float4 s[NSAMP];
#pragma unroll
  for (int i = 0; i < NSAMP; ++i) {
    s[i] = reinterpret_cast<const float4*>(samples + (size_t)i * NPIX)[t];
  }

  // term1: sum_i |s_i - y|
  float4 a = make_float4(0.f, 0.f, 0.f, 0.f);
#pragma unroll
  for (int i = 0; i < NSAMP; ++i) {
    a.x += fabsf(s[i].x - y.x);
    a.y += fabsf(s[i].y - y.y);
    a.z += fabsf(s[i].z - y.z);
    a.w += fabsf(s[i].w - y.w);
  }

  // pairwise: sum_{i<j} |s_i - s_j|  (120 pairs, fully in registers)
  float4 b = make_float4(0.f, 0.f, 0.f, 0.f);
#pragma unroll
  for (int i = 1; i < NSAMP; ++i) {
#pragma unroll
    for (int j = 0; j < i; ++j) {
      b.x += fabsf(s[i].x - s[j].x);
      b.y += fabsf(s[i].y - s[j].y);
      b.z += fabsf(s[i].z - s[j].z);
      b.w += fabsf(s[i].w - s[j].w);
    }
  }

  // per-thread contribution over its 4 pixels:
  //   (1/N)*term1_sum - (1/N^2)*pairwise_sum   (final 1/NPIX applied in pass 2)
  float c = (a.x + a.y + a.z + a.w) * (1.0f / 16.0f) -
            (b.x + b.y + b.z + b.w) * (1.0f / 256.0f);

  // per-wave reduction on the matrix pipe (no divergence up to here -> EXEC full)
  float wsum = wave_sum64_wmma(c, 0.0f);

  __shared__ float red[NTHREADS / 32];
  const int lane = threadIdx.x & 31;
  const int wid  = threadIdx.x >> 5;
  if (lane == 0) red[wid] = wsum;
  __syncthreads();
  if (threadIdx.x == 0) {
    float r = 0.f;
#pragma unroll
    for (int k = 0; k < NTHREADS / 32; ++k) r += red[k];
    partials[blockIdx.x] = r;
  }
}

__global__ void crps_final_kernel(const float* __restrict__ partials,
                                  float* __restrict__ out) {
  const int t = threadIdx.x;  // single wave32, EXEC all ones, uniform trip count
  float acc0 = 0.f, acc1 = 0.f;
#pragma unroll
  for (int k = 0; k < NBLOCKS / 64; ++k) {
    acc0 += partials[t + 64 * k];
    acc1 += partials[t + 32 + 64 * k];
  }
  float total = wave_sum64_wmma(acc0, acc1);
  if (t == 0) out[0] = total * (1.0f / (float)NPIX);
}

extern "C" void kernel_launch(void* const* d_in, const int* in_sizes, int n_in,
                              void* d_out, int out_size, void* d_ws, size_t ws_size,
                              hipStream_t stream) {
  const float* samples = (const float*)d_in[0];   // [16, 262144] f32
  const float* target  = (const float*)d_in[1];   // [262144] f32
  float* out      = (float*)d_out;                // 1 f32
  float* partials = (float*)d_ws;                 // 256 f32 scratch

  crps_partial_kernel<<<NBLOCKS, NTHREADS, 0, stream>>>(samples, target, partials);
  crps_final_kernel<<<1, 32, 0, stream>>>(partials, out);
}